// GNN_12412455486167
// MI455X (gfx1250) — compile-verified
//
#include <hip/hip_runtime.h>

typedef float v2f __attribute__((ext_vector_type(2)));
typedef float v8f __attribute__((ext_vector_type(8)));

// ---------------------------------------------------------------------------
// Elementwise helpers
// ---------------------------------------------------------------------------
__global__ void zero_f32(float* __restrict__ p, int n) {
  int i = blockIdx.x * blockDim.x + threadIdx.x;
  if (i < n) p[i] = 0.0f;
}

__global__ void degree_kernel(const int* __restrict__ src, const int* __restrict__ dst,
                              float* __restrict__ deg_out, float* __restrict__ deg_in,
                              int n_edges) {
  int e = blockIdx.x * blockDim.x + threadIdx.x;
  if (e >= n_edges) return;
  atomicAdd(deg_out + src[e], 1.0f);
  atomicAdd(deg_in + dst[e], 1.0f);
}

__global__ void norm_kernel(float* __restrict__ deg, int n) {
  int i = blockIdx.x * blockDim.x + threadIdx.x;
  if (i >= n) return;
  deg[i] = rsqrtf(fmaxf(deg[i], 1.0f));
}

// out[i] = agg[i] * norm_d[row] + bias[col], optional ReLU
__global__ void post_kernel(const float* __restrict__ agg, const float* __restrict__ norm_d,
                            const float* __restrict__ bias, float* __restrict__ out,
                            int n_elems, int do_relu) {
  int i = blockIdx.x * blockDim.x + threadIdx.x;
  if (i >= n_elems) return;
  int row = i >> 6;
  int col = i & 63;
  float v = fmaf(agg[i], norm_d[row], bias[col]);
  if (do_relu) v = fmaxf(v, 0.0f);
  out[i] = v;
}

// ---------------------------------------------------------------------------
// Edge scatter-add SpMM: agg[dst[e], :] += H[src[e], :]   (64 cols)
// 16 lanes per edge, float4 per lane, f32 atomics (L2-resident: H fits in 192MB L2)
// ---------------------------------------------------------------------------
__global__ void spmm_scatter(const float* __restrict__ H, const int* __restrict__ src,
                             const int* __restrict__ dst, float* __restrict__ agg,
                             int n_edges) {
  int t = blockIdx.x * blockDim.x + threadIdx.x;
  int e = t >> 4;
  if (e >= n_edges) return;
  int c = (t & 15) * 4;
  int s = src[e];
  int d = dst[e];
  const float4 v = *reinterpret_cast<const float4*>(H + (size_t)s * 64 + c);
  float* o = agg + (size_t)d * 64 + c;
  atomicAdd(o + 0, v.x);
  atomicAdd(o + 1, v.y);
  atomicAdd(o + 2, v.z);
  atomicAdd(o + 3, v.w);
}

// ---------------------------------------------------------------------------
// Y[m, 0:64] = (X[m, 0:K] * norm_s[m]) @ W[K, 64]  via V_WMMA_F32_16X16X4_F32
// One wave per 16-row tile, 4 accumulators cover N=64. W staged in LDS.
// n_nodes % 16 == 0 so active waves run full tiles (EXEC all ones for WMMA).
// ---------------------------------------------------------------------------
template <int K>
__global__ void __launch_bounds__(256) gemm_norm_wmma(const float* __restrict__ X,
                                                      const float* __restrict__ W,
                                                      const float* __restrict__ norm_s,
                                                      float* __restrict__ Y,
                                                      int n_nodes) {
  __shared__ float sW[K * 64];
  for (int i = threadIdx.x; i < K * 64; i += 256) sW[i] = W[i];
  __syncthreads();

  const int wave = threadIdx.x >> 5;
  const int lane = threadIdx.x & 31;
  const int row0 = (blockIdx.x * 8 + wave) * 16;
  if (row0 + 15 >= n_nodes) return;  // whole-wave uniform; no further barriers

  const int half  = lane >> 4;        // 0: K offsets {0,1}; 1: {2,3}
  const int sub   = lane & 15;        // M index for A, N index for B/C
  const int mrow  = row0 + sub;
  const float ns  = norm_s[mrow];
  const float* __restrict__ xrow = X + (size_t)mrow * K;

  v8f acc0 = {}, acc1 = {}, acc2 = {}, acc3 = {};

#pragma unroll
  for (int kk = 0; kk < K / 4; ++kk) {
    const int ka = kk * 4 + half * 2;  // this lane's K pair within the 16x4 A tile
    v2f a;
    a.x = xrow[ka] * ns;
    a.y = xrow[ka + 1] * ns;

    const float* wk = &sW[ka * 64 + sub];  // B: rows ka, ka+1; col = n0 + sub
    v2f b;

    b.x = wk[0];       b.y = wk[64];
    acc0 = __builtin_amdgcn_wmma_f32_16x16x4_f32(false, a, false, b, (short)0, acc0, false, false);
    b.x = wk[16];      b.y = wk[64 + 16];
    acc1 = __builtin_amdgcn_wmma_f32_16x16x4_f32(false, a, false, b, (short)0, acc1, false, false);
    b.x = wk[32];      b.y = wk[64 + 32];
    acc2 = __builtin_amdgcn_wmma_f32_16x16x4_f32(false, a, false, b, (short)0, acc2, false, false);
    b.x = wk[48];      b.y = wk[64 + 48];
    acc3 = __builtin_amdgcn_wmma_f32_16x16x4_f32(false, a, false, b, (short)0, acc3, false, false);
  }

  // C/D layout: VGPR r -> M = r + 8*half, N = sub (per 16x16 f32 table)
  float* ybase = Y + (size_t)row0 * 64 + sub;
#pragma unroll
  for (int r = 0; r < 8; ++r) {
    const int m = r + half * 8;
    float* yr = ybase + (size_t)m * 64;
    yr[0]  = acc0[r];
    yr[16] = acc1[r];
    yr[32] = acc2[r];
    yr[48] = acc3[r];
  }
}

// ---------------------------------------------------------------------------
// Launch
// ---------------------------------------------------------------------------
static inline int cdiv(long long a, long long b) { return (int)((a + b - 1) / b); }

extern "C" void kernel_launch(void* const* d_in, const int* in_sizes, int n_in,
                              void* d_out, int out_size, void* d_ws, size_t ws_size,
                              hipStream_t stream) {
  const float* features = (const float*)d_in[0];
  const int*   src      = (const int*)d_in[1];
  const int*   dst      = (const int*)d_in[2];
  const float* W1       = (const float*)d_in[3];
  const float* b1       = (const float*)d_in[4];
  const float* W2       = (const float*)d_in[5];
  const float* b2       = (const float*)d_in[6];

  const int n_nodes = in_sizes[0] / 128;
  const int n_edges = in_sizes[1];
  const int n_feat  = n_nodes * 64;

  float* ws     = (float*)d_ws;
  float* norm_s = ws;                               // also deg_out
  float* norm_d = ws + n_nodes;                     // also deg_in
  float* bufA   = ws + 2 * (size_t)n_nodes;         // GEMM output / layer-1 activations
  float* bufB   = bufA + (size_t)n_nodes * 64;      // aggregation / GEMM-2 output
  float* out    = (float*)d_out;

  const int B = 256;

  // degrees -> norms
  zero_f32<<<cdiv(2 * (long long)n_nodes, B), B, 0, stream>>>(norm_s, 2 * n_nodes);
  degree_kernel<<<cdiv(n_edges, B), B, 0, stream>>>(src, dst, norm_s, norm_d, n_edges);
  norm_kernel<<<cdiv(2 * (long long)n_nodes, B), B, 0, stream>>>(norm_s, 2 * n_nodes);

  const int gemm_blocks = cdiv(n_nodes, 128);  // 8 waves x 16 rows per block

  // ---- layer 1 ----
  gemm_norm_wmma<128><<<gemm_blocks, B, 0, stream>>>(features, W1, norm_s, bufA, n_nodes);
  zero_f32<<<cdiv(n_feat, B), B, 0, stream>>>(bufB, n_feat);
  spmm_scatter<<<cdiv((long long)n_edges * 16, B), B, 0, stream>>>(bufA, src, dst, bufB, n_edges);
  post_kernel<<<cdiv(n_feat, B), B, 0, stream>>>(bufB, norm_d, b1, bufA, n_feat, 1);

  // ---- layer 2 ----
  gemm_norm_wmma<64><<<gemm_blocks, B, 0, stream>>>(bufA, W2, norm_s, bufB, n_nodes);
  zero_f32<<<cdiv(n_feat, B), B, 0, stream>>>(out, n_feat);
  spmm_scatter<<<cdiv((long long)n_edges * 16, B), B, 0, stream>>>(bufB, src, dst, out, n_edges);
  post_kernel<<<cdiv(n_feat, B), B, 0, stream>>>(out, norm_d, b2, out, n_feat, 0);
}